// MultiHeadRoPE_28166395528043
// MI455X (gfx1250) — compile-verified
//
#include <hip/hip_runtime.h>

typedef float v2f __attribute__((ext_vector_type(2)));
typedef float v4f __attribute__((ext_vector_type(4)));
typedef float v8f __attribute__((ext_vector_type(8)));

#define SEQ   4096
#define DIM   64
#define NT    (SEQ / 16)   // k-tiles
#define KSTR  68           // padded K/V LDS row stride (bank = 4*row+col mod 64)
#define SSTR  18           // padded S LDS row stride

__global__ __launch_bounds__(256) void MultiHeadRoPE_28166395528043_kernel(
    const float* __restrict__ q,  const float* __restrict__ k,  const float* __restrict__ v,
    const float* __restrict__ fq, const float* __restrict__ fk, const float* __restrict__ fv,
    float* __restrict__ out)
{
    __shared__ float Kl[2][16][KSTR];
    __shared__ float Vl[2][16][KSTR];
    __shared__ float Sl[8][16][SSTR];

    const int batch = blockIdx.x >> 5;          // SEQ/128 = 32 q-blocks per batch
    const int q0    = (blockIdx.x & 31) * 128;
    const int tid   = threadIdx.x;
    const int wave  = tid >> 5;
    const int lane  = tid & 31;
    const int lm    = lane & 15;                // M / N index within 16
    const int lh    = lane >> 4;                // half-wave select (K-pair offset)

    const float scale = 0.125f;                 // 1/sqrt(64)

    const float* qb = q + (size_t)batch * SEQ * DIM;
    const float* kb = k + (size_t)batch * SEQ * DIM;
    const float* vb = v + (size_t)batch * SEQ * DIM;
    float*       ob = out + (size_t)batch * SEQ * DIM;

    // ---- RoPE Q straight into WMMA A-fragment layout (16x4 f32 chunks) ----
    // lane L: row = L%16, K-pair = 4c + 2*(L/16) .. +1  (pair == RoPE pair)
    v2f qa[16];
    {
        const int    qrow = q0 + wave * 16 + lm;
        const float* qp   = qb + (size_t)qrow * DIM;
        const float* fp   = fq + (size_t)qrow * DIM;   // (cos,sin) interleaved
        #pragma unroll
        for (int c = 0; c < 16; ++c) {
            const int col = c * 4 + 2 * lh;
            v2f x = *(const v2f*)(qp + col);
            v2f f = *(const v2f*)(fp + col);
            v2f r;
            r.x = x.x * f.x - x.y * f.y;
            r.y = x.x * f.y + x.y * f.x;
            qa[c] = r;
        }
    }

    v8f acc0 = {}, acc1 = {}, acc2 = {}, acc3 = {};   // 16x64 output tile

    // cooperative tile loader mapping: 256 threads x 4 floats = one 16x64 tile
    const int lrow = tid >> 4;        // 0..15
    const int lcol = (tid & 15) * 4;  // 0,4,...,60

    // ---- preload k-tile 0 (RoPE applied in registers) ----
    {
        v4f xk  = *(const v4f*)(kb + (size_t)lrow * DIM + lcol);
        v4f fk4 = *(const v4f*)(fk + (size_t)lrow * DIM + lcol);
        v4f xv  = *(const v4f*)(vb + (size_t)lrow * DIM + lcol);
        v4f fv4 = *(const v4f*)(fv + (size_t)lrow * DIM + lcol);
        v4f rk, rv;
        rk.x = xk.x*fk4.x - xk.y*fk4.y;  rk.y = xk.x*fk4.y + xk.y*fk4.x;
        rk.z = xk.z*fk4.z - xk.w*fk4.w;  rk.w = xk.z*fk4.w + xk.w*fk4.z;
        rv.x = xv.x*fv4.x - xv.y*fv4.y;  rv.y = xv.x*fv4.y + xv.y*fv4.x;
        rv.z = xv.z*fv4.z - xv.w*fv4.w;  rv.w = xv.z*fv4.w + xv.w*fv4.z;
        *(v4f*)&Kl[0][lrow][lcol] = rk;
        *(v4f*)&Vl[0][lrow][lcol] = rv;
    }

    for (int kt = 0; kt < NT; ++kt) {
        __syncthreads();
        const int buf = kt & 1;

        // ---- double-buffered load of next tile (overlaps with WMMA below) ----
        if (kt + 1 < NT) {
            const int nb   = buf ^ 1;
            const int krow = (kt + 1) * 16 + lrow;
            if (kt + 2 < NT) {   // warm L2/WGP$ two tiles ahead
                __builtin_prefetch(kb + (size_t)(krow + 16) * DIM + lcol, 0, 1);
                __builtin_prefetch(vb + (size_t)(krow + 16) * DIM + lcol, 0, 1);
            }
            v4f xk  = *(const v4f*)(kb + (size_t)krow * DIM + lcol);
            v4f fk4 = *(const v4f*)(fk + (size_t)krow * DIM + lcol);
            v4f xv  = *(const v4f*)(vb + (size_t)krow * DIM + lcol);
            v4f fv4 = *(const v4f*)(fv + (size_t)krow * DIM + lcol);
            v4f rk, rv;
            rk.x = xk.x*fk4.x - xk.y*fk4.y;  rk.y = xk.x*fk4.y + xk.y*fk4.x;
            rk.z = xk.z*fk4.z - xk.w*fk4.w;  rk.w = xk.z*fk4.w + xk.w*fk4.z;
            rv.x = xv.x*fv4.x - xv.y*fv4.y;  rv.y = xv.x*fv4.y + xv.y*fv4.x;
            rv.z = xv.z*fv4.z - xv.w*fv4.w;  rv.w = xv.z*fv4.w + xv.w*fv4.z;
            *(v4f*)&Kl[nb][lrow][lcol] = rk;
            *(v4f*)&Vl[nb][lrow][lcol] = rv;
        }

        // ---- S (16x16) = Q_tile (16x64) x K_tileT : 16 chained f32 WMMAs ----
        v8f s = {};
        #pragma unroll
        for (int c = 0; c < 16; ++c) {
            v2f bfrag = *(const v2f*)&Kl[buf][lm][c * 4 + 2 * lh];
            s = __builtin_amdgcn_wmma_f32_16x16x4_f32(
                    false, qa[c], false, bfrag, (short)0, s, false, false);
        }

        // ---- scale + park S in per-wave LDS (D-layout -> A-layout relayout) ----
        // D layout: element r of lane L is S[r + 8*(L/16)][L%16]
        #pragma unroll
        for (int r = 0; r < 8; ++r)
            Sl[wave][r + 8 * lh][lm] = s[r] * scale;
        // DS ops are in-order within a wave: the gathers below see these stores.

        // ---- out (16x64) += S_scaled (16x16) x V_tile (16x64) ----
        #pragma unroll
        for (int c2 = 0; c2 < 4; ++c2) {
            v2f a2 = *(const v2f*)&Sl[wave][lm][c2 * 4 + 2 * lh];
            const int vr = c2 * 4 + 2 * lh;
            v2f b0, b1, b2, b3;
            b0.x = Vl[buf][vr][ 0 + lm];  b0.y = Vl[buf][vr + 1][ 0 + lm];
            b1.x = Vl[buf][vr][16 + lm];  b1.y = Vl[buf][vr + 1][16 + lm];
            b2.x = Vl[buf][vr][32 + lm];  b2.y = Vl[buf][vr + 1][32 + lm];
            b3.x = Vl[buf][vr][48 + lm];  b3.y = Vl[buf][vr + 1][48 + lm];
            acc0 = __builtin_amdgcn_wmma_f32_16x16x4_f32(false, a2, false, b0, (short)0, acc0, false, false);
            acc1 = __builtin_amdgcn_wmma_f32_16x16x4_f32(false, a2, false, b1, (short)0, acc1, false, false);
            acc2 = __builtin_amdgcn_wmma_f32_16x16x4_f32(false, a2, false, b2, (short)0, acc2, false, false);
            acc3 = __builtin_amdgcn_wmma_f32_16x16x4_f32(false, a2, false, b3, (short)0, acc3, false, false);
        }
    }

    // ---- store output tile (D-layout: row = r + 8*lh, col = 16*db + lm) ----
    const int orow = q0 + wave * 16;
    #pragma unroll
    for (int r = 0; r < 8; ++r) {
        const size_t row = (size_t)(orow + r + 8 * lh) * DIM;
        ob[row +  0 + lm] = acc0[r];
        ob[row + 16 + lm] = acc1[r];
        ob[row + 32 + lm] = acc2[r];
        ob[row + 48 + lm] = acc3[r];
    }
}

extern "C" void kernel_launch(void* const* d_in, const int* in_sizes, int n_in,
                              void* d_out, int out_size, void* d_ws, size_t ws_size,
                              hipStream_t stream) {
    const float* q  = (const float*)d_in[0];
    const float* k  = (const float*)d_in[1];
    const float* v  = (const float*)d_in[2];
    const float* fq = (const float*)d_in[3];
    const float* fk = (const float*)d_in[4];
    const float* fv = (const float*)d_in[5];
    float* out = (float*)d_out;

    dim3 grid(8 * (SEQ / 128));   // 8 batches x 32 q-blocks = 256 WGs
    dim3 block(256);              // 8 wave32s
    hipLaunchKernelGGL(MultiHeadRoPE_28166395528043_kernel, grid, block, 0, stream,
                       q, k, v, fq, fk, fv, out);
}